// GNN_12558484374183
// MI455X (gfx1250) — compile-verified
//
#include <hip/hip_runtime.h>

// ---------------------------------------------------------------------------
// GNN forward on MI455X (gfx1250, wave32).
//   L1: agg1 = scatter_add(x[src]) ; h1 = relu(agg1@W1_rel.T + b1 + x@W1_root.T)
//   L2: agg2 = scatter_add(h1[src]); h2 = relu(agg2@W2_rel.T + b2 + h1@W2_root.T)
//   out = mean(h2,0) @ Wlin.T + blin          (h2 never materialized; pooled fused)
// Dense layers use V_WMMA_F32_16X16X4_F32 (f32 in, f32 acc = reference precision).
// ---------------------------------------------------------------------------

typedef __attribute__((ext_vector_type(2))) float f32x2;
typedef __attribute__((ext_vector_type(8))) float f32x8;

#define WMMA_F32X4(a, b, c) \
    __builtin_amdgcn_wmma_f32_16x16x4_f32(false, (a), false, (b), (short)0, (c), false, false)

// ---------------- Layer-1 aggregation: agg1[dst] += x[src], Fin=4 -----------
__global__ void __launch_bounds__(256) gnn_l1_agg(
    const int* __restrict__ esrc, const int* __restrict__ edst,
    const float* __restrict__ x, float* __restrict__ agg1, int E)
{
    int e = blockIdx.x * 256 + threadIdx.x;
    if (e >= E) return;
    int s = esrc[e], d = edst[e];
    float4 v = *(const float4*)(x + (size_t)s * 4);
    float* p = agg1 + (size_t)d * 4;
    unsafeAtomicAdd(p + 0, v.x);
    unsafeAtomicAdd(p + 1, v.y);
    unsafeAtomicAdd(p + 2, v.z);
    unsafeAtomicAdd(p + 3, v.w);
}

// ---------------- Layer-1 dense (WMMA f32 16x16x4, K=4 exactly) -------------
// Block = 256 threads = 8 waves. Block rb owns node rows [rb*16, rb*16+16).
// Wave w owns output columns [w*16, w*16+16). Two WMMAs per wave (rel + root).
__global__ void __launch_bounds__(256) gnn_l1_dense(
    const float* __restrict__ agg1, const float* __restrict__ x,
    const float* __restrict__ W1_rel, const float* __restrict__ b1,
    const float* __restrict__ W1_root, float* __restrict__ h1)
{
    const int lane = threadIdx.x & 31;
    const int wv   = threadIdx.x >> 5;          // column tile 0..7
    const int rb   = blockIdx.x;                // row block
    const int m    = lane & 15;                 // A-row / B-col within tile
    const int koff = (lane < 16) ? 0 : 2;       // K half (ISA f32 A/B layout)
    const int row  = rb * 16 + m;
    const int col  = wv * 16 + m;

    // A tiles: 16x4 (nodes x Fin), lane holds float2 at K = koff..koff+1
    f32x2 a_rel  = *(const f32x2*)(agg1 + (size_t)row * 4 + koff);
    f32x2 a_root = *(const f32x2*)(x    + (size_t)row * 4 + koff);
    // B tiles: B[k][n] = W[n][k]; W is [128,4] row-major -> contiguous float2
    f32x2 b_rel  = *(const f32x2*)(W1_rel  + (size_t)col * 4 + koff);
    f32x2 b_root = *(const f32x2*)(W1_root + (size_t)col * 4 + koff);

    f32x8 c = {};
    c = WMMA_F32X4(a_rel,  b_rel,  c);
    c = WMMA_F32X4(a_root, b_root, c);

    const float bias  = b1[col];
    const int   rbase = rb * 16 + ((lane >= 16) ? 8 : 0);
#pragma unroll
    for (int v = 0; v < 8; ++v) {
        float val = c[v] + bias;
        val = val > 0.f ? val : 0.f;            // relu
        h1[(size_t)(rbase + v) * 128 + col] = val;
    }
}

// ---------------- Layer-2 aggregation: agg2[dst] += h1[src], F=128 ----------
// One wave per edge; lane L moves features [4L, 4L+4) -> fully coalesced 512B.
// Edge index forced wave-uniform (readfirstlane) so src/dst come from scalar
// loads and the gather/scatter use SGPR-base addressing.
__global__ void __launch_bounds__(256) gnn_l2_agg(
    const int* __restrict__ esrc, const int* __restrict__ edst,
    const float* __restrict__ h1, float* __restrict__ agg2, int E)
{
    const int lane = threadIdx.x & 31;
    int e = blockIdx.x * 8 + (threadIdx.x >> 5);
    if (e >= E) return;                       // wave-uniform exit
    e = __builtin_amdgcn_readfirstlane(e);
    const int s = __builtin_amdgcn_readfirstlane(esrc[e]);
    const int d = __builtin_amdgcn_readfirstlane(edst[e]);
    float4 v = *(const float4*)(h1 + (size_t)s * 128 + lane * 4);
    float* p = agg2 + (size_t)d * 128 + lane * 4;
    unsafeAtomicAdd(p + 0, v.x);
    unsafeAtomicAdd(p + 1, v.y);
    unsafeAtomicAdd(p + 2, v.z);
    unsafeAtomicAdd(p + 3, v.w);
}

// ---------------- Layer-2 dense + fused mean-pool ---------------------------
// Block rb: stage agg2/h1 16x128 tiles in LDS (stride 132 -> no bank conflicts).
// Wave w: K-loop of 32 steps x 2 GEMMs = 64 f32 WMMAs, two independent
// accumulator chains (rel / root) interleaved to relax XDL RAW chaining.
// Then relu + column sums -> one atomic per column per block (h2 never stored).
#define LSTR 132
__global__ void __launch_bounds__(256) gnn_l2_dense_pool(
    const float* __restrict__ agg2, const float* __restrict__ h1,
    const float* __restrict__ W2_rel, const float* __restrict__ b2,
    const float* __restrict__ W2_root, float* __restrict__ pooled)
{
    __shared__ float sA[16 * LSTR];   // agg2 tile
    __shared__ float sH[16 * LSTR];   // h1 tile
    const int rb = blockIdx.x;
    const int t  = threadIdx.x;

    {   // cooperative tile load: 256 threads x 8 floats (2x float4), coalesced
        const int r  = t >> 4;
        const int cb = (t & 15) * 8;
        const float* gA = agg2 + ((size_t)rb * 16 + r) * 128 + cb;
        const float* gH = h1   + ((size_t)rb * 16 + r) * 128 + cb;
        *(float4*)(sA + r * LSTR + cb)     = *(const float4*)(gA);
        *(float4*)(sA + r * LSTR + cb + 4) = *(const float4*)(gA + 4);
        *(float4*)(sH + r * LSTR + cb)     = *(const float4*)(gH);
        *(float4*)(sH + r * LSTR + cb + 4) = *(const float4*)(gH + 4);
    }
    __syncthreads();

    const int lane = t & 31;
    const int wv   = t >> 5;
    const int m    = lane & 15;
    const int koff = (lane < 16) ? 0 : 2;
    const int col  = wv * 16 + m;

    const float* wrel  = W2_rel  + (size_t)col * 128;  // B[k][n] = W[n][k]
    const float* wroot = W2_root + (size_t)col * 128;

    f32x8 c0 = {};   // agg2 @ W2_rel.T chain
    f32x8 c1 = {};   // h1   @ W2_root.T chain
#pragma unroll
    for (int kk = 0; kk < 32; ++kk) {
        const int kb = kk * 4 + koff;
        f32x2 ar = *(const f32x2*)(sA + m * LSTR + kb);
        f32x2 br = *(const f32x2*)(wrel + kb);
        c0 = WMMA_F32X4(ar, br, c0);
        f32x2 ah = *(const f32x2*)(sH + m * LSTR + kb);
        f32x2 bh = *(const f32x2*)(wroot + kb);
        c1 = WMMA_F32X4(ah, bh, c1);
    }

    const float bias = b2[col];
    float s = 0.f;
#pragma unroll
    for (int v = 0; v < 8; ++v) {
        float val = c0[v] + c1[v] + bias;
        val = val > 0.f ? val : 0.f;            // relu(h2)
        s += val;                                // partial column sum (8 rows)
    }
    s += __shfl_xor(s, 16, 32);                  // combine M halves (wave32)
    if (lane < 16) unsafeAtomicAdd(&pooled[col], s);
}

// ---------------- Final: out = (pooled/N) @ Wlin.T + blin -------------------
__global__ void __launch_bounds__(64) gnn_final(
    const float* __restrict__ pooled, const float* __restrict__ Wlin,
    const float* __restrict__ blin, float* __restrict__ out, float invN)
{
    const int lane = threadIdx.x & 31;
    const int i    = threadIdx.x >> 5;           // output 0 or 1
    float s = 0.f;
#pragma unroll
    for (int j = lane; j < 128; j += 32)
        s += pooled[j] * invN * Wlin[i * 128 + j];
#pragma unroll
    for (int off = 16; off > 0; off >>= 1)
        s += __shfl_xor(s, off, 32);
    if (lane == 0) out[i] = s + blin[i];
}

// ---------------------------------------------------------------------------
extern "C" void kernel_launch(void* const* d_in, const int* in_sizes, int n_in,
                              void* d_out, int out_size, void* d_ws, size_t ws_size,
                              hipStream_t stream)
{
    const float* x       = (const float*)d_in[0];
    const int*   edge    = (const int*)  d_in[1];
    const float* W1_rel  = (const float*)d_in[2];
    const float* b1      = (const float*)d_in[3];
    const float* W1_root = (const float*)d_in[4];
    const float* W2_rel  = (const float*)d_in[5];
    const float* b2      = (const float*)d_in[6];
    const float* W2_root = (const float*)d_in[7];
    const float* Wlin    = (const float*)d_in[8];
    const float* blin    = (const float*)d_in[9];
    float* out = (float*)d_out;

    const int N = in_sizes[0] / 4;      // 100000 (multiple of 16)
    const int E = in_sizes[1] / 2;      // 1600000
    const int* esrc = edge;
    const int* edst = edge + E;

    // workspace layout (all 256B-aligned given these sizes)
    char*  ws     = (char*)d_ws;
    float* agg1   = (float*)ws;                                   // N*4
    float* h1     = (float*)(ws + (size_t)N * 4 * 4);             // N*128
    float* agg2   = (float*)(ws + (size_t)N * 4 * 4 + (size_t)N * 128 * 4);
    float* pooled = (float*)(ws + (size_t)N * 4 * 4 + (size_t)N * 128 * 4 * 2);

    // zero accumulators every call (d_ws is not re-poisoned between replays)
    hipMemsetAsync(agg1,   0, (size_t)N * 4   * sizeof(float), stream);
    hipMemsetAsync(agg2,   0, (size_t)N * 128 * sizeof(float), stream);
    hipMemsetAsync(pooled, 0, 128 * sizeof(float), stream);

    const int nrb = N / 16;             // 6250 row blocks

    gnn_l1_agg<<<(E + 255) / 256, 256, 0, stream>>>(esrc, edst, x, agg1, E);
    gnn_l1_dense<<<nrb, 256, 0, stream>>>(agg1, x, W1_rel, b1, W1_root, h1);
    gnn_l2_agg<<<(E + 7) / 8, 256, 0, stream>>>(esrc, edst, h1, agg2, E);
    gnn_l2_dense_pool<<<nrb, 256, 0, stream>>>(agg2, h1, W2_rel, b2, W2_root, pooled);
    gnn_final<<<1, 64, 0, stream>>>(pooled, Wlin, blin, out, 1.0f / (float)N);
}